// FPSDownsample_592705487236
// MI455X (gfx1250) — compile-verified
//
#include <hip/hip_runtime.h>

#define NPTS 131072
#define NSAMP 1024
#define BATCH 32
#define FPS_TPB 1024
#define PPT (NPTS / FPS_TPB)   // 128 points per thread

typedef float v2f __attribute__((ext_vector_type(2)));
typedef float v8f __attribute__((ext_vector_type(8)));

#if defined(__HIP_DEVICE_COMPILE__) && __has_builtin(__builtin_amdgcn_wmma_f32_16x16x4_f32)
#define HAVE_WMMA_F32 1
#else
#define HAVE_WMMA_F32 0
#endif

// ---------------------------------------------------------------------------
// Farthest point sampling: one workgroup per batch (32 waves, wave32).
// dist[] lives in global scratch (L2-resident); each thread privately owns a
// stride-1024 slice of points -> coalesced, race-free min-update.
// ---------------------------------------------------------------------------
__global__ __launch_bounds__(FPS_TPB) void fps_kernel(
    const float* __restrict__ x,        // [B, N, 3]
    const int*   __restrict__ init_c,   // [B]
    float*       __restrict__ dist,     // [B, N] scratch
    float*       __restrict__ sampled)  // [B, S, 3] (front of d_out)
{
    const int b   = blockIdx.x;
    const int tid = threadIdx.x;
    const float* __restrict__ xb = x + (size_t)b * NPTS * 3;
    float* __restrict__ db = dist + (size_t)b * NPTS;

    // re-init distances every call (ws is not re-poisoned between replays)
    #pragma unroll 4
    for (int j = 0; j < PPT; ++j) db[tid + j * FPS_TPB] = 1e10f;

    __shared__ float s_val[32];
    __shared__ int   s_idx[32];
    __shared__ float s_c[3];
    __shared__ int   s_ci;

    if (tid == 0) s_ci = init_c[b];
    __syncthreads();
    int cur = s_ci;

    for (int t = 0; t < NSAMP; ++t) {
        if (tid == 0) {
            const float cx = xb[cur * 3 + 0];
            const float cy = xb[cur * 3 + 1];
            const float cz = xb[cur * 3 + 2];
            s_c[0] = cx; s_c[1] = cy; s_c[2] = cz;
            float* sp = sampled + ((size_t)b * NSAMP + t) * 3;  // record CURRENT centroid
            sp[0] = cx; sp[1] = cy; sp[2] = cz;
        }
        __syncthreads();
        const float cx = s_c[0], cy = s_c[1], cz = s_c[2];

        float best = -1.0f;
        int   bi   = 0x7fffffff;
        #pragma unroll 4
        for (int j = 0; j < PPT; ++j) {
            const int i = tid + j * FPS_TPB;
            const float px = xb[i * 3 + 0];
            const float py = xb[i * 3 + 1];
            const float pz = xb[i * 3 + 2];
            const float dx = px - cx, dy = py - cy, dz = pz - cz;
            float d = fmaf(dx, dx, fmaf(dy, dy, dz * dz));
            d = fminf(db[i], d);
            db[i] = d;
            // strict > keeps the lowest index on ties (argmax semantics)
            if (d > best) { best = d; bi = i; }
        }
        // wave32 butterfly reduction (value, index), prefer lower index on ties
        #pragma unroll
        for (int off = 16; off > 0; off >>= 1) {
            const float ov = __shfl_down(best, off, 32);
            const int   oi = __shfl_down(bi,   off, 32);
            if (ov > best || (ov == best && oi < bi)) { best = ov; bi = oi; }
        }
        const int wid = tid >> 5;
        if ((tid & 31) == 0) { s_val[wid] = best; s_idx[wid] = bi; }
        __syncthreads();
        if (tid < 32) {
            best = s_val[tid];
            bi   = s_idx[tid];
            #pragma unroll
            for (int off = 16; off > 0; off >>= 1) {
                const float ov = __shfl_down(best, off, 32);
                const int   oi = __shfl_down(bi,   off, 32);
                if (ov > best || (ov == best && oi < bi)) { best = ov; bi = oi; }
            }
            if (tid == 0) s_ci = bi;
        }
        __syncthreads();
        cur = s_ci;
    }
}

// ---------------------------------------------------------------------------
// FP32 GEMM + bias (+ReLU) via V_WMMA_F32_16X16X4_F32.
// One wave per 16x16 output tile. EXEC stays all-ones (wave-uniform guard,
// exact grids). K is zero-padded to a multiple of 4 in registers (layer 1).
// A 16x4 layout: lanes 0-15 -> K=k,k+1 ; lanes 16-31 -> K=k+2,k+3.
// C/D 16x16:     VGPR r -> row m0+r (lanes 0-15) / m0+r+8 (lanes 16-31).
// ---------------------------------------------------------------------------
template <int K, int O, bool RELU>
__global__ __launch_bounds__(256) void mlp_gemm(
    const float* __restrict__ IN,    // [M, K]
    const float* __restrict__ W,     // [K, O]
    const float* __restrict__ bias,  // [O]
    float*       __restrict__ OUT,   // [M, O]
    int ntiles)
{
    const int lane = threadIdx.x & 31;
    const int wave = threadIdx.x >> 5;
    const int w    = blockIdx.x * (blockDim.x >> 5) + wave;
    if (w >= ntiles) return;                 // wave-uniform: EXEC stays full

    constexpr int OT = O / 16;
    const int rt = w / OT;
    const int ct = w % OT;
    const int hlf = lane >> 4;               // 0: lanes 0-15, 1: lanes 16-31
    const int l   = lane & 15;
    const int m0  = rt * 16;
    const int col = ct * 16 + l;
    constexpr int KP = (K + 3) & ~3;

#if HAVE_WMMA_F32
    v8f c = {};
    #pragma unroll
    for (int k = 0; k < KP; k += 4) {
        const int kk  = k + hlf * 2;
        const int row = m0 + l;
        v2f a, bm;
        a.x  = (kk     < K) ? IN[row * K + kk]       : 0.0f;
        a.y  = (kk + 1 < K) ? IN[row * K + kk + 1]   : 0.0f;
        bm.x = (kk     < K) ? W[kk * O + col]        : 0.0f;
        bm.y = (kk + 1 < K) ? W[(kk + 1) * O + col]  : 0.0f;
        c = __builtin_amdgcn_wmma_f32_16x16x4_f32(
            /*neg_a=*/false, a, /*neg_b=*/false, bm,
            /*c_mod=*/(short)0, c, /*reuse_a=*/false, /*reuse_b=*/false);
    }
    const float bv = bias[col];
    #pragma unroll
    for (int r = 0; r < 8; ++r) {
        const int row = m0 + r + 8 * hlf;
        float v = c[r] + bv;
        if (RELU) v = fmaxf(v, 0.0f);
        OUT[row * O + col] = v;
    }
#else
    // scalar fallback with identical output mapping
    const float bv = bias[col];
    #pragma unroll
    for (int r = 0; r < 8; ++r) {
        const int row = m0 + r + 8 * hlf;
        float acc = 0.0f;
        for (int kk = 0; kk < K; ++kk) acc = fmaf(IN[row * K + kk], W[kk * O + col], acc);
        float v = acc + bv;
        if (RELU) v = fmaxf(v, 0.0f);
        OUT[row * O + col] = v;
    }
#endif
}

// ---------------------------------------------------------------------------
extern "C" void kernel_launch(void* const* d_in, const int* in_sizes, int n_in,
                              void* d_out, int out_size, void* d_ws, size_t ws_size,
                              hipStream_t stream) {
    const float* x     = (const float*)d_in[0];
    const int*   initc = (const int*)  d_in[1];
    const float* W1    = (const float*)d_in[2];
    const float* b1    = (const float*)d_in[3];
    const float* W2    = (const float*)d_in[4];
    const float* b2    = (const float*)d_in[5];
    const float* W3    = (const float*)d_in[6];
    const float* b3    = (const float*)d_in[7];

    float* out     = (float*)d_out;
    float* sampled = out;                                   // [32,1024,3]
    float* feats   = out + (size_t)BATCH * NSAMP * 3;       // [32,1024,256]

    const size_t M = (size_t)BATCH * NSAMP;                 // 32768 rows
    char* ws = (char*)d_ws;
    float* H1   = (float*)ws;                               // [M,64]   8 MB
    float* H2   = (float*)(ws + M * 64 * sizeof(float));    // [M,128] 16 MB
    float* dist = (float*)(ws + M * 64 * sizeof(float)
                              + M * 128 * sizeof(float));   // [B,N]   16 MB

    // 1) FPS: one block per batch
    fps_kernel<<<BATCH, FPS_TPB, 0, stream>>>(x, initc, dist, sampled);

    // 2) MLP: 16x16 tile per wave, 8 waves per block (exact grids)
    const int t1 = (int)(M / 16) * (64 / 16);    // 8192  tiles
    const int t2 = (int)(M / 16) * (128 / 16);   // 16384 tiles
    const int t3 = (int)(M / 16) * (256 / 16);   // 32768 tiles
    mlp_gemm<3,   64,  true ><<<t1 / 8, 256, 0, stream>>>(sampled, W1, b1, H1,   t1);
    mlp_gemm<64,  128, true ><<<t2 / 8, 256, 0, stream>>>(H1,      W2, b2, H2,   t2);
    mlp_gemm<128, 256, false><<<t3 / 8, 256, 0, stream>>>(H2,      W3, b3, feats, t3);
}